// HopfieldLayer_35699768165228
// MI455X (gfx1250) — compile-verified
//
#include <hip/hip_runtime.h>
#include <math.h>
#include <stdint.h>

typedef __attribute__((ext_vector_type(2))) float v2f;
typedef __attribute__((ext_vector_type(8))) float v8f;
typedef __attribute__((ext_vector_type(4))) unsigned int v4u;
typedef __attribute__((ext_vector_type(4))) int v4i;
typedef __attribute__((ext_vector_type(8))) int v8i;

#define B_ROWS 131072
#define DDIM   64
#define P      68            // padded LDS row pitch in floats (8B aligned b64 slots, bank-friendly)
#define WPB    4             // waves per block (wave32)
#define RPW    16            // rows per wave (one 16xN WMMA tile)
#define RPB    (WPB*RPW)     // 64 rows per block
#define NBLK   (B_ROWS/RPB)  // 2048 blocks
#define WPITCH 160           // LDS pitch (floats) per W k-pair: lane halves hit disjoint bank halves

// workspace byte layout
#define WS_SC    0                       // float: [0]=t [1]=dt
#define WS_IC    64                      // int:   [0]=done [1]=accept
#define WS_WP    256                     // Wpair: 4096 floats (kpair*128 + 2*col + p)
#define WS_ERRP  (256 + 4096*4)          // 2048 floats
#define WS_VELP  (WS_ERRP + 2048*4)      // 2048 floats
#define WS_X5    (WS_VELP + 2048*4)      // 131072*64 floats (33024 is 256B aligned)

#define HAVE_TDM __has_builtin(__builtin_amdgcn_tensor_load_to_lds)

// ---------------------------------------------------------------------------
// Tensor Data Mover: 2-D tile load global->LDS with LDS row padding.
// tile_w elems/row (fp32), tile_h rows, row stride in elems; pad codes per D#:
// pad_interval code: 1<<(code+1) dwords between pads, pad_amount code: (code+1) dwords.
// ---------------------------------------------------------------------------
#if HAVE_TDM
__device__ __forceinline__ void tdm_load_2d(unsigned lds_byte_addr, const void* gsrc,
                                            unsigned tile_w, unsigned tile_h,
                                            unsigned row_stride_elems,
                                            unsigned pad_int_code, unsigned pad_amt_code)
{
    unsigned long long ga = (unsigned long long)(uintptr_t)gsrc;
    v4u g0;
    g0.x = 1u;                                       // count=1, user descriptor
    g0.y = lds_byte_addr;                            // lds_addr @ bits 63:32
    g0.z = (unsigned)(ga & 0xffffffffu);             // global_addr[31:0] @ 95:64
    g0.w = (unsigned)((ga >> 32) & 0x01ffffffu)      // global_addr[56:32] @ 120:96
         | (2u << 30);                               // type=2 ("image")
    v8i g1;
    g1[0] = (int)((2u << 16)                         // data_size = 4B
                | (1u << 20)                         // pad_enable
                | (pad_int_code << 22)
                | (pad_amt_code << 25));
    g1[1] = (int)((tile_w & 0xffffu) << 16);         // tensor_dim0[15:0] @ 63:48
    g1[2] = (int)(((tile_w >> 16) & 0xffffu)         // tensor_dim0[31:16]
                | ((tile_h & 0xffffu) << 16));       // tensor_dim1[15:0]
    g1[3] = (int)(((tile_h >> 16) & 0xffffu)         // tensor_dim1[31:16]
                | ((tile_w & 0xffffu) << 16));       // tile_dim0 @ 127:112
    g1[4] = (int)(tile_h & 0xffffu);                 // tile_dim1 @ 143:128 (tile_dim2=0)
    g1[5] = (int)row_stride_elems;                   // tensor_dim0_stride[31:0]
    g1[6] = 0;
    g1[7] = 0;
    v4i gz = {0, 0, 0, 0};                           // groups 2/3: dims 2..4 unused
#if __clang_major__ >= 23
    v8i z8 = {0, 0, 0, 0, 0, 0, 0, 0};
    __builtin_amdgcn_tensor_load_to_lds(g0, g1, gz, gz, z8, 0);
#else
    __builtin_amdgcn_tensor_load_to_lds(g0, g1, gz, gz, 0);
#endif
}

__device__ __forceinline__ unsigned lds_off(const void* p)
{
    return (unsigned)(uintptr_t)p;   // generic LDS addr: low 32 bits = LDS byte offset
}

__device__ __forceinline__ void wait_tensor0(void)
{
#if __has_builtin(__builtin_amdgcn_s_wait_tensorcnt)
    __builtin_amdgcn_s_wait_tensorcnt(0);
#else
    asm volatile("s_wait_tensorcnt 0x0" ::: "memory");
#endif
}
#endif

// ---------------------------------------------------------------------------
// f(x) = sqrt(max(x*(1-x),0)) * (-(x @ Ws) - b) for 16 rows per wave.
// xs: LDS, 16 rows, pitch P.  kout: LDS (or nullptr).  Accumulates c5*f, c4*f
// into C-fragment-shaped register arrays.
// ---------------------------------------------------------------------------
__device__ __forceinline__ void f_eval(const float* xs, float* kout,
                                       const float* sW, const float* sb,
                                       float c5, float c4,
                                       float s5[4][8], float s4[4][8])
{
    const int lane = threadIdx.x & 31;
    const int half = lane >> 4;
    const int r0   = lane & 15;
#pragma unroll
    for (int nt = 0; nt < 4; ++nt) {
        const int col = nt * 16 + r0;
        const float bc = sb[col];
        v8f acc = {bc, bc, bc, bc, bc, bc, bc, bc};   // start at +b so grad = -acc
#pragma unroll
        for (int kt = 0; kt < 16; ++kt) {
            // A 16x4 fp32 frag: lanes<16 -> K=4kt+0,1 ; lanes>=16 -> K=4kt+2,3
            v2f a = *(const v2f*)&xs[r0 * P + 4 * kt + 2 * half];
            // B 4x16 fp32 frag: (Ws[K][col], Ws[K+1][col]) pair-interleaved in LDS
            v2f w = *(const v2f*)&sW[(2 * kt + half) * WPITCH + 2 * col];
            acc = __builtin_amdgcn_wmma_f32_16x16x4_f32(false, a, false, w,
                                                        (short)0, acc, false, false);
        }
#pragma unroll
        for (int v = 0; v < 8; ++v) {
            const int rr = v + 8 * half;                 // C/D layout: M = v (+8 for hi lanes)
            float xe = xs[rr * P + col];
            float fe = sqrtf(fmaxf(xe * (1.0f - xe), 0.0f)) * (-acc[v]);
            if (kout) kout[rr * P + col] = fe;
            s5[nt][v] += c5 * fe;
            s4[nt][v] += c4 * fe;
        }
    }
}

// ---------------------------------------------------------------------------
__global__ void rkf_init(const float* __restrict__ x0, const float* __restrict__ W,
                         float* __restrict__ X, float* __restrict__ Wp,
                         float* sc, int* ic)
{
    size_t gid = (size_t)blockIdx.x * blockDim.x + threadIdx.x;
    const size_t n4 = (size_t)B_ROWS * DDIM / 4;
    if (gid < n4) ((float4*)X)[gid] = ((const float4*)x0)[gid];
    if (gid < 4096) {
        int kp = (int)gid >> 7, j = (int)gid & 127;
        int c = j >> 1, p = j & 1;
        int k = 2 * kp + p;
        Wp[gid] = 0.5f * (W[k * 64 + c] + W[c * 64 + k]);   // symmetrized Ws
    }
    if (gid == 0) { sc[0] = 0.0f; sc[1] = 0.5f; ic[0] = 0; ic[1] = 0; }
}

// ---------------------------------------------------------------------------
// One full RKF45 step (6 fused stages) for 64 rows; writes x5 and err partial.
// ---------------------------------------------------------------------------
__global__ __launch_bounds__(128) void rkf_stepA(const float* __restrict__ X,
                                                 float* __restrict__ X5,
                                                 const float* __restrict__ Wp_g,
                                                 const float* __restrict__ b_g,
                                                 const float* sc, const int* ic,
                                                 float* __restrict__ errP)
{
    __shared__ float sW[32 * WPITCH];
    __shared__ float sb[64];
    __shared__ float sRed[128];
    __shared__ float sBuf[WPB][7][RPW * P];   // [x, k1..k5, xs] per wave

    if (ic[0]) return;
    const float dt = sc[1];
    const int t = threadIdx.x;
    const int w = t >> 5, lane = t & 31;
    const int rowbase = blockIdx.x * RPB;

#if HAVE_TDM
    // TDM: per-wave 16x64 fp32 tile of X, padded to LDS pitch 68 (64 dw + 4 dw pad)
    tdm_load_2d(lds_off(&sBuf[w][0][0]), X + (size_t)(rowbase + w * RPW) * DDIM,
                64u, 16u, 64u, /*pad: every 64 dw*/ 5u, /*+4 dw*/ 3u);
    if (w == 0)   // Wpair 32x128 -> LDS pitch 160 (128 dw + 32 dw pad)
        tdm_load_2d(lds_off(&sW[0]), Wp_g, 128u, 32u, 128u, /*every 128 dw*/ 6u, /*+32 dw*/ 31u);
    if (t < 64) sb[t] = b_g[t];
    wait_tensor0();
#else
    for (int i = t; i < 4096; i += 128) { int kp = i >> 7, j = i & 127; sW[kp * WPITCH + j] = Wp_g[i]; }
    if (t < 64) sb[t] = b_g[t];
    for (int i = 0; i < 32; ++i) {
        int e = t + 128 * i;
        int row = e >> 6, col = e & 63;
        sBuf[row >> 4][0][(row & 15) * P + col] = X[(size_t)(rowbase + row) * DDIM + col];
    }
#endif
    __syncthreads();

    float* xb = sBuf[w][0];
    float* k1 = sBuf[w][1]; float* k2 = sBuf[w][2]; float* k3 = sBuf[w][3];
    float* k4 = sBuf[w][4]; float* k5 = sBuf[w][5]; float* xs = sBuf[w][6];

    float s5[4][8], s4[4][8];
#pragma unroll
    for (int n = 0; n < 4; ++n)
#pragma unroll
        for (int v = 0; v < 8; ++v) { s5[n][v] = 0.0f; s4[n][v] = 0.0f; }

    // stage 1
    f_eval(xb, k1, sW, sb, 16.0f/135.0f, 25.0f/216.0f, s5, s4);
    __syncthreads();
    // stage 2
    for (int i = 0; i < 32; ++i) { int e = i*32 + lane; int id = (e>>6)*P + (e&63);
        xs[id] = xb[id] + dt * (0.25f * k1[id]); }
    __syncthreads();
    f_eval(xs, k2, sW, sb, 0.0f, 0.0f, s5, s4);
    __syncthreads();
    // stage 3
    for (int i = 0; i < 32; ++i) { int e = i*32 + lane; int id = (e>>6)*P + (e&63);
        xs[id] = xb[id] + dt * ((3.0f/32.0f)*k1[id] + (9.0f/32.0f)*k2[id]); }
    __syncthreads();
    f_eval(xs, k3, sW, sb, 6656.0f/12825.0f, 1408.0f/2565.0f, s5, s4);
    __syncthreads();
    // stage 4
    for (int i = 0; i < 32; ++i) { int e = i*32 + lane; int id = (e>>6)*P + (e&63);
        xs[id] = xb[id] + dt * ((1932.0f/2197.0f)*k1[id] - (7200.0f/2197.0f)*k2[id]
                              + (7296.0f/2197.0f)*k3[id]); }
    __syncthreads();
    f_eval(xs, k4, sW, sb, 28561.0f/56430.0f, 2197.0f/4104.0f, s5, s4);
    __syncthreads();
    // stage 5
    for (int i = 0; i < 32; ++i) { int e = i*32 + lane; int id = (e>>6)*P + (e&63);
        xs[id] = xb[id] + dt * ((439.0f/216.0f)*k1[id] - 8.0f*k2[id]
                              + (3680.0f/513.0f)*k3[id] - (845.0f/4104.0f)*k4[id]); }
    __syncthreads();
    f_eval(xs, k5, sW, sb, -9.0f/50.0f, -0.2f, s5, s4);
    __syncthreads();
    // stage 6 (k6 only feeds s5)
    for (int i = 0; i < 32; ++i) { int e = i*32 + lane; int id = (e>>6)*P + (e&63);
        xs[id] = xb[id] + dt * ((-8.0f/27.0f)*k1[id] + 2.0f*k2[id]
                              - (3544.0f/2565.0f)*k3[id] + (1859.0f/4104.0f)*k4[id]
                              - (11.0f/40.0f)*k5[id]); }
    __syncthreads();
    f_eval(xs, nullptr, sW, sb, 2.0f/55.0f, 0.0f, s5, s4);

    // x5 out + err = max |x5 - x4| = dt * max |s5 - s4|
    const int half = lane >> 4, r0 = lane & 15;
    float emax = 0.0f;
#pragma unroll
    for (int nt = 0; nt < 4; ++nt) {
        const int col = nt * 16 + r0;
#pragma unroll
        for (int v = 0; v < 8; ++v) {
            int rr = v + 8 * half;
            float xe = xb[rr * P + col];
            X5[(size_t)(rowbase + w * 16 + rr) * DDIM + col] = xe + dt * s5[nt][v];
            emax = fmaxf(emax, fabsf(dt * (s5[nt][v] - s4[nt][v])));
        }
    }
    sRed[t] = emax;
    __syncthreads();
    if (t == 0) { float m = 0.0f; for (int i = 0; i < 128; ++i) m = fmaxf(m, sRed[i]);
                  errP[blockIdx.x] = m; }
}

// ---------------------------------------------------------------------------
// err -> accept, t, dt   (single block, deterministic)
// ---------------------------------------------------------------------------
__global__ __launch_bounds__(256) void rkf_ctrl1(float* sc, int* ic, const float* __restrict__ errP)
{
    __shared__ float red[256];
    if (ic[0]) return;
    int t = threadIdx.x;
    float m = 0.0f;
    for (int i = t; i < NBLK; i += 256) m = fmaxf(m, errP[i]);
    red[t] = m;
    __syncthreads();
    if (t == 0) {
        float err = 0.0f;
        for (int i = 0; i < 256; ++i) err = fmaxf(err, red[i]);
        float dt = sc[1];
        int acc = (err <= 0.01f) || (dt <= 0.1f);
        if (acc) sc[0] = sc[0] + dt;
        float fac = 0.9f * powf(0.01f / fmaxf(err, 1e-12f), 0.2f);
        fac = fminf(fmaxf(fac, 0.2f), 4.0f);
        dt = fminf(fmaxf(dt * fac, 0.1f), 10.0f);
        sc[1] = dt;
        ic[1] = acc;
    }
}

// ---------------------------------------------------------------------------
// x_n = accept ? x5 : x ; write back; vel partial = sum over rows of max|f(x_n)|
// (also used for the initial vel0 pass with accept==0)
// ---------------------------------------------------------------------------
__global__ __launch_bounds__(128) void rkf_stepB(float* __restrict__ X,
                                                 const float* __restrict__ X5,
                                                 const float* __restrict__ Wp_g,
                                                 const float* __restrict__ b_g,
                                                 const int* ic, float* __restrict__ velP)
{
    __shared__ float sW[32 * WPITCH];
    __shared__ float sb[64];
    __shared__ float sRed[RPB];
    __shared__ float sBuf[WPB][2][RPW * P];   // [x_n, f] per wave

    if (ic[0]) return;
    const int accept = ic[1];
    const int t = threadIdx.x;
    const int w = t >> 5, lane = t & 31;
    const int rowbase = blockIdx.x * RPB;
    const float* src = accept ? X5 : X;

#if HAVE_TDM
    tdm_load_2d(lds_off(&sBuf[w][0][0]), src + (size_t)(rowbase + w * RPW) * DDIM,
                64u, 16u, 64u, 5u, 3u);
    if (w == 0)
        tdm_load_2d(lds_off(&sW[0]), Wp_g, 128u, 32u, 128u, 6u, 31u);
    if (t < 64) sb[t] = b_g[t];
    wait_tensor0();
    __syncthreads();
    if (accept) {           // write accepted state back, 8B vectors from LDS
        for (int i = 0; i < 16; ++i) {
            int i2 = t + 128 * i;               // 0..2047 float2s
            int row = i2 >> 5, c2 = i2 & 31;
            v2f val = *(const v2f*)&sBuf[row >> 4][0][(row & 15) * P + 2 * c2];
            *(v2f*)&X[(size_t)(rowbase + row) * DDIM + 2 * c2] = val;
        }
    }
#else
    for (int i = t; i < 4096; i += 128) { int kp = i >> 7, j = i & 127; sW[kp * WPITCH + j] = Wp_g[i]; }
    if (t < 64) sb[t] = b_g[t];
    for (int i = 0; i < 32; ++i) {
        int e = t + 128 * i;
        int row = e >> 6, col = e & 63;
        size_t g = (size_t)(rowbase + row) * DDIM + col;
        float v = src[g];
        sBuf[row >> 4][0][(row & 15) * P + col] = v;
        if (accept) X[g] = v;
    }
    __syncthreads();
#endif

    float s5[4][8], s4[4][8];
#pragma unroll
    for (int n = 0; n < 4; ++n)
#pragma unroll
        for (int v = 0; v < 8; ++v) { s5[n][v] = 0.0f; s4[n][v] = 0.0f; }
    f_eval(sBuf[w][0], sBuf[w][1], sW, sb, 0.0f, 0.0f, s5, s4);
    __syncthreads();

    if (lane < 16) {
        float m = 0.0f;
        const float* kr = &sBuf[w][1][lane * P];
        for (int c = 0; c < 64; ++c) m = fmaxf(m, fabsf(kr[c]));
        sRed[w * 16 + lane] = m;
    }
    __syncthreads();
    if (t == 0) { float s = 0.0f; for (int i = 0; i < RPB; ++i) s += sRed[i];
                  velP[blockIdx.x] = s; }
}

// ---------------------------------------------------------------------------
// vel -> done   (single block, deterministic)
// ---------------------------------------------------------------------------
__global__ __launch_bounds__(256) void rkf_ctrl2(const float* sc, int* ic, const float* __restrict__ velP)
{
    __shared__ float red[256];
    if (ic[0]) return;
    int t = threadIdx.x;
    float s = 0.0f;
    for (int i = t; i < NBLK; i += 256) s += velP[i];
    red[t] = s;
    __syncthreads();
    if (t == 0) {
        float tot = 0.0f;
        for (int i = 0; i < 256; ++i) tot += red[i];
        float vel = tot / (float)B_ROWS;
        if ((sc[0] > 50.0f) || (vel < 0.01f)) ic[0] = 1;
    }
}

// ---------------------------------------------------------------------------
extern "C" void kernel_launch(void* const* d_in, const int* in_sizes, int n_in,
                              void* d_out, int out_size, void* d_ws, size_t ws_size,
                              hipStream_t stream)
{
    const float* x0 = (const float*)d_in[0];
    const float* W  = (const float*)d_in[1];
    const float* b  = (const float*)d_in[2];
    float* X = (float*)d_out;                 // working state doubles as output
    char*  ws = (char*)d_ws;
    float* sc   = (float*)(ws + WS_SC);
    int*   ic   = (int*)  (ws + WS_IC);
    float* Wp   = (float*)(ws + WS_WP);
    float* errP = (float*)(ws + WS_ERRP);
    float* velP = (float*)(ws + WS_VELP);
    float* X5   = (float*)(ws + WS_X5);

    rkf_init<<<(B_ROWS * DDIM / 4 + 255) / 256, 256, 0, stream>>>(x0, W, X, Wp, sc, ic);
    // initial velocity / done (accept==0 -> evaluates f at x0, no writeback)
    rkf_stepB<<<NBLK, 128, 0, stream>>>(X, X5, Wp, b, ic, velP);
    rkf_ctrl2<<<1, 256, 0, stream>>>(sc, ic, velP);
    for (int it = 0; it < 16; ++it) {
        rkf_stepA<<<NBLK, 128, 0, stream>>>(X, X5, Wp, b, sc, ic, errP);
        rkf_ctrl1<<<1, 256, 0, stream>>>(sc, ic, errP);
        rkf_stepB<<<NBLK, 128, 0, stream>>>(X, X5, Wp, b, ic, velP);
        rkf_ctrl2<<<1, 256, 0, stream>>>(sc, ic, velP);
    }
}